// CoverageMechanism_37125697306812
// MI455X (gfx1250) — compile-verified
//
#include <hip/hip_runtime.h>

// Repetition-penalty kernel for MI455X (gfx1250).
// out[b,i,v] = logits[b,i,v] - 0.3 * (# of tokens[b, i-4..i-1] equal to v), i>=4
// Pure HBM-streaming problem (524 MB traffic, ~22.5us floor @ 23.3 TB/s).
// Data path: async global->LDS (ASYNCcnt) double-buffered per wave, penalty
// applied in-register, non-temporal 128-bit stores.

#define VOCAB   32000
#define SEQ     512
#define PEN_W   0.3f
#define ROW_F4  (VOCAB / 4)                 // 8000 float4 per row
#define TPB     256                         // 8 wave32 per block
#define NIT     ((ROW_F4 + TPB - 1) / TPB)  // 32 iterations (last partial)

typedef float f4 __attribute__((ext_vector_type(4)));
typedef int   i4 __attribute__((ext_vector_type(4)));

#if defined(__gfx1250__) && __has_builtin(__builtin_amdgcn_global_load_async_to_lds_b128)
#define ASYNC_OK 1
#else
#define ASYNC_OK 0
#endif

#if __has_builtin(__builtin_amdgcn_s_wait_asynccnt)
#define WAIT_ASYNC(n) __builtin_amdgcn_s_wait_asynccnt(n)
#else
#define WAIT_ASYNC(n) asm volatile("s_wait_asynccnt %0" ::"i"(n) : "memory")
#endif

#if ASYNC_OK
__device__ __forceinline__ void async_cp16(f4* lds_dst, const f4* g_src) {
  // global_load_async_to_lds_b128: per-lane 16B global -> LDS, tracked by ASYNCcnt.
  // Toolchain signature wants int4-vector pointers (per round-1 diagnostic).
  __builtin_amdgcn_global_load_async_to_lds_b128(
      (i4*)g_src, (i4*)lds_dst, /*imm offset=*/0, /*cpol=*/0);
}
#endif

__global__ __launch_bounds__(TPB) void rep_penalty_kernel(
    const float* __restrict__ logits, const int* __restrict__ toks,
    float* __restrict__ out) {
  const int row = blockIdx.x;      // row = b*SEQ + i, one block per row
  const int b   = row >> 9;        // SEQ == 512
  const int i   = row & (SEQ - 1);
  const long rowOff = (long)row * VOCAB;
  const f4* __restrict__ src = (const f4*)(logits + rowOff);
  f4* __restrict__ dst       = (f4*)(out + rowOff);

  // Window tokens (uniform per block -> scalar loads). -1 never matches v>=0.
  int t0 = -1, t1 = -1, t2 = -1, t3 = -1;
  if (i >= 4) {
    const int* tw = toks + b * SEQ + (i - 4);
    t0 = tw[0]; t1 = tw[1]; t2 = tw[2]; t3 = tw[3];
  }

  const int tid = threadIdx.x;

#if ASYNC_OK
  // Double-buffered async pipeline. Each wave only reads the LDS lanes it
  // wrote itself, so s_wait_asynccnt (per-wave, in-order loads) is the only
  // synchronization needed -- no workgroup barriers.
  __shared__ f4 buf[2][TPB];

  // Prologue: stage chunk 0 (tid < 8000 always).
  async_cp16(&buf[0][tid], src + tid);

  for (int it = 0; it < NIT; ++it) {
    const int stage = it & 1;
    if (it + 1 < NIT) {
      // Clamp the tail address instead of masking EXEC: every wave issues
      // exactly one async op per iteration, keeping ASYNCcnt uniform.
      int nidx = (it + 1) * TPB + tid;
      nidx = nidx < ROW_F4 ? nidx : (ROW_F4 - 1);
      async_cp16(&buf[stage ^ 1][tid], src + nidx);
      WAIT_ASYNC(1);  // oldest (current chunk) complete; prefetch in flight
    } else {
      WAIT_ASYNC(0);  // drain
    }
    asm volatile("" ::: "memory");

    f4 v = buf[stage][tid];                 // ds_load_b128
    const int idx = it * TPB + tid;
    const int v0  = idx << 2;
#pragma unroll
    for (int c = 0; c < 4; ++c) {
      const int vv  = v0 + c;
      const int cnt = (vv == t0) + (vv == t1) + (vv == t2) + (vv == t3);
      v[c] -= PEN_W * (float)cnt;
    }
    if (idx < ROW_F4)
      __builtin_nontemporal_store(v, dst + idx);  // global_store_b128 th:NT
  }
#else
  // Fallback: direct 128-bit streaming (still the roofline-correct shape).
  for (int idx = tid; idx < ROW_F4; idx += TPB) {
    f4 v = __builtin_nontemporal_load(src + idx);
    const int v0 = idx << 2;
#pragma unroll
    for (int c = 0; c < 4; ++c) {
      const int vv  = v0 + c;
      const int cnt = (vv == t0) + (vv == t1) + (vv == t2) + (vv == t3);
      v[c] -= PEN_W * (float)cnt;
    }
    __builtin_nontemporal_store(v, dst + idx);
  }
#endif
}

extern "C" void kernel_launch(void* const* d_in, const int* in_sizes, int n_in,
                              void* d_out, int out_size, void* d_ws, size_t ws_size,
                              hipStream_t stream) {
  const float* logits = (const float*)d_in[0];   // [B, S, V] f32
  const int*   toks   = (const int*)d_in[1];     // [B, S] int
  float*       out    = (float*)d_out;           // [B, S, V] f32
  const int rows = in_sizes[1];                  // B * S (= 2048)
  rep_penalty_kernel<<<rows, TPB, 0, stream>>>(logits, toks, out);
}